// EncoderDecoderModel_86792699117972
// MI455X (gfx1250) — compile-verified
//
#include <hip/hip_runtime.h>
#include <math.h>

// Problem sizes (fixed by the reference)
#define BB 8
#define SS 128
#define TT 128
#define VV 32000
#define EE 512
#define HH 512
#define GG 2048   // 4*H

typedef __attribute__((ext_vector_type(2))) float v2f;
typedef __attribute__((ext_vector_type(8))) float v8f;

__device__ __forceinline__ v8f wmma4(v2f a, v2f b, v8f c) {
  // V_WMMA_F32_16X16X4_F32 : D = A(16x4) x B(4x16) + C(16x16), all fp32
  return __builtin_amdgcn_wmma_f32_16x16x4_f32(false, a, false, b, (short)0, c,
                                               false, false);
}
__device__ __forceinline__ v2f ld2(const float* p) { return *(const v2f*)p; }
__device__ __forceinline__ v8f zero8() {
  v8f z = {0.f, 0.f, 0.f, 0.f, 0.f, 0.f, 0.f, 0.f};
  return z;
}
__device__ __forceinline__ float sigm(float x) { return 1.0f / (1.0f + expf(-x)); }

// ---------------------------------------------------------------------------
// Init: zero h double-buffer (2*16*H), c (16*H) and the 2 grid-sync counters.
// Rows 8..15 of h/c stay zero forever (M-padding for the 16-row WMMA tiles).
// ---------------------------------------------------------------------------
__global__ void k_init(float* hc, int n, unsigned* cnt) {
  int i = blockIdx.x * blockDim.x + threadIdx.x;
  if (i < n) hc[i] = 0.0f;
  if (i < 2) cnt[i] = 0u;
}

// ---------------------------------------------------------------------------
// Embedding gather, time-major output: xt[(t*B + b)*E + e] = emb[idx[b,t]]
// ---------------------------------------------------------------------------
__global__ void k_gather(const int* __restrict__ idx, const float* __restrict__ emb,
                         float* __restrict__ xt, int L) {
  long i = (long)blockIdx.x * blockDim.x + threadIdx.x;  // L*B*E threads
  int e = (int)(i % EE);
  long r = i / EE;
  int t = (int)(r / BB);
  int b = (int)(r % BB);
  xt[((long)(t * BB + b)) * EE + e] = emb[(long)idx[b * L + t] * EE + e];
}

// ---------------------------------------------------------------------------
// Xg = x_tm @ Wih^T + (bih + bhh).  M=1024 (t*B+b), K=E=512, N=4H=2048.
// Wave tile: 16 rows x 64 cols (1 M-tile x 4 N-tiles). 2048 waves.
// ---------------------------------------------------------------------------
__global__ void k_xg_gemm(const float* __restrict__ xt, const float* __restrict__ Wih,
                          const float* __restrict__ bih, const float* __restrict__ bhh,
                          float* __restrict__ Xg) {
  const int lane = threadIdx.x & 31;
  const int wid = blockIdx.x * (blockDim.x >> 5) + (threadIdx.x >> 5);
  const int mt = wid >> 5;   // 0..63  (M tile of 16)
  const int ng = wid & 31;   // 0..31  (group of 4 N tiles = 64 cols)
  const int lm = lane & 15, kh = lane >> 4;

  v8f acc[4];
#pragma unroll
  for (int j = 0; j < 4; ++j) acc[j] = zero8();

  const float* arow = xt + (size_t)(mt * 16 + lm) * EE + 2 * kh;
  const float* brow[4];
#pragma unroll
  for (int j = 0; j < 4; ++j)
    brow[j] = Wih + (size_t)(ng * 64 + j * 16 + lm) * EE + 2 * kh;

  for (int kb = 0; kb < EE; kb += 4) {
    v2f a = ld2(arow + kb);
#pragma unroll
    for (int j = 0; j < 4; ++j) acc[j] = wmma4(a, ld2(brow[j] + kb), acc[j]);
  }

#pragma unroll
  for (int j = 0; j < 4; ++j) {
    int n = ng * 64 + j * 16 + lm;
    float bias = bih[n] + bhh[n];
#pragma unroll
    for (int v = 0; v < 8; ++v) {
      int m = mt * 16 + v + 8 * kh;
      Xg[(size_t)m * GG + n] = acc[j][v] + bias;
    }
  }
}

// ---------------------------------------------------------------------------
// Persistent LSTM recurrence. 8 blocks x 128 threads = 32 waves; wave w owns
// h columns [16w,16w+16) and computes its i/f/g/o gate tiles (K=512) with
// f32 WMMA each step. h is double-buffered; one grid barrier per step.
// outseq layout: [B, L, H].
// ---------------------------------------------------------------------------
__global__ void k_lstm(const float* __restrict__ Xg, const float* __restrict__ Whh,
                       float* __restrict__ hbuf, float* __restrict__ cbuf,
                       float* __restrict__ outseq, unsigned* __restrict__ cnt, int L) {
  const int lane = threadIdx.x & 31;
  const int w = blockIdx.x * (blockDim.x >> 5) + (threadIdx.x >> 5);  // 0..31
  const int lm = lane & 15, kh = lane >> 4;
  const int nb = w * 16;
  const unsigned NB = gridDim.x;

  const float* brow[4];
#pragma unroll
  for (int g = 0; g < 4; ++g)
    brow[g] = Whh + (size_t)(g * HH + nb + lm) * HH + 2 * kh;

  for (int t = 0; t < L; ++t) {
    const float* hread = hbuf + (size_t)(t & 1) * 16 * HH;
    float* hwrite = hbuf + (size_t)((t + 1) & 1) * 16 * HH;

    v8f acc[4];
#pragma unroll
    for (int g = 0; g < 4; ++g) acc[g] = zero8();

    const float* arow = hread + (size_t)lm * HH + 2 * kh;
    for (int kb = 0; kb < HH; kb += 4) {
      v2f a = ld2(arow + kb);
#pragma unroll
      for (int g = 0; g < 4; ++g) acc[g] = wmma4(a, ld2(brow[g] + kb), acc[g]);
    }

    if (lane < 16) {  // lanes 0..15 hold batch rows 0..7 in acc[...][0..7]
      int n = nb + lm;
#pragma unroll
      for (int v = 0; v < 8; ++v) {
        size_t row = (size_t)(t * BB + v) * GG;
        float iv = acc[0][v] + Xg[row + 0 * HH + n];
        float fv = acc[1][v] + Xg[row + 1 * HH + n];
        float gv = acc[2][v] + Xg[row + 2 * HH + n];
        float ov = acc[3][v] + Xg[row + 3 * HH + n];
        float co = cbuf[(size_t)v * HH + n];
        float cn = sigm(fv) * co + sigm(iv) * tanhf(gv);
        float hn = sigm(ov) * tanhf(cn);
        cbuf[(size_t)v * HH + n] = cn;
        hwrite[(size_t)v * HH + n] = hn;
        outseq[((size_t)v * L + t) * HH + n] = hn;
      }
    }

    // Grid-wide barrier. Release side: EVERY thread fences so each wave
    // flushes/waits its own outstanding h-stores (STOREcnt is per-wave;
    // s_barrier alone gives no device-scope visibility). Then one thread
    // per block arrives and spins. Acquire side: all threads fence after
    // the barrier so every wave invalidates before reading the new h.
    __threadfence();
    __syncthreads();
    if (threadIdx.x == 0) {
      atomicAdd(cnt, 1u);
      unsigned target = (unsigned)(t + 1) * NB;
      while (__hip_atomic_load(cnt, __ATOMIC_RELAXED, __HIP_MEMORY_SCOPE_AGENT) <
             target) {
      }
    }
    __syncthreads();
    __threadfence();
  }
}

// ---------------------------------------------------------------------------
// Additive attention + concat. One block per (b,t); 128 threads.
// score[s] = sum_h scale[h]*tanh(q[h]+k[s,h]); softmax over s; context;
// concat[b*T+t] = [context(512) | dec_out(512)]
// ---------------------------------------------------------------------------
__global__ void k_attn(const float* __restrict__ enc_out, const float* __restrict__ dec_out,
                       const float* __restrict__ scale, float* __restrict__ concat) {
  __shared__ float q[HH];
  __shared__ float sc[HH];
  __shared__ float wgt[SS];
  __shared__ float red[128];
  const int b = blockIdx.x / TT, t = blockIdx.x % TT;
  const int i = threadIdx.x;  // 0..127

  for (int h = i; h < HH; h += 128) {
    q[h] = dec_out[((size_t)b * TT + t) * HH + h];
    sc[h] = scale[h];
  }
  __syncthreads();

  const float* er = enc_out + ((size_t)b * SS + i) * HH;
  float acc = 0.f;
  for (int h = 0; h < HH; ++h) acc += sc[h] * tanhf(q[h] + er[h]);

  red[i] = acc;
  __syncthreads();
  for (int off = 64; off > 0; off >>= 1) {
    if (i < off) red[i] = fmaxf(red[i], red[i + off]);
    __syncthreads();
  }
  float mx = red[0];
  __syncthreads();
  float ex = expf(acc - mx);
  red[i] = ex;
  __syncthreads();
  for (int off = 64; off > 0; off >>= 1) {
    if (i < off) red[i] += red[i + off];
    __syncthreads();
  }
  float inv = 1.0f / red[0];
  __syncthreads();
  wgt[i] = ex * inv;
  __syncthreads();

  float* crow = concat + ((size_t)b * TT + t) * (2 * HH);
  for (int h = i; h < HH; h += 128) {
    float ctx = 0.f;
    for (int s = 0; s < SS; ++s) ctx += wgt[s] * enc_out[((size_t)b * SS + s) * HH + h];
    crow[h] = ctx;
    crow[HH + h] = q[h];
  }
}

// ---------------------------------------------------------------------------
// logits = concat[1024,1024] @ out_W^T[1024,32000] + out_b.
// Wave tile: 64x64 (4 M-tiles x 4 N-tiles), K=1024, f32 WMMA.
// 8000 waves -> 2000 blocks of 128 threads. out: [B*T, V] fp32.
// ---------------------------------------------------------------------------
__global__ void k_out_gemm(const float* __restrict__ concat, const float* __restrict__ W,
                           const float* __restrict__ bias, float* __restrict__ out) {
  const int lane = threadIdx.x & 31;
  const int wid = blockIdx.x * (blockDim.x >> 5) + (threadIdx.x >> 5);  // 0..7999
  const int mb = wid / 500;  // 0..15 (64-row block)
  const int nbk = wid % 500; // 0..499 (64-col block)
  const int lm = lane & 15, kh = lane >> 4;
  const int K = 2 * HH;

  v8f acc[16];
#pragma unroll
  for (int x = 0; x < 16; ++x) acc[x] = zero8();

  const float* ar[4];
  const float* br[4];
#pragma unroll
  for (int ii = 0; ii < 4; ++ii)
    ar[ii] = concat + (size_t)(mb * 64 + ii * 16 + lm) * K + 2 * kh;
#pragma unroll
  for (int j = 0; j < 4; ++j)
    br[j] = W + (size_t)(nbk * 64 + j * 16 + lm) * K + 2 * kh;

  for (int kb = 0; kb < K; kb += 4) {
    v2f a0 = ld2(ar[0] + kb), a1 = ld2(ar[1] + kb);
    v2f a2 = ld2(ar[2] + kb), a3 = ld2(ar[3] + kb);
#pragma unroll
    for (int j = 0; j < 4; ++j) {
      v2f bv = ld2(br[j] + kb);
      acc[0 * 4 + j] = wmma4(a0, bv, acc[0 * 4 + j]);
      acc[1 * 4 + j] = wmma4(a1, bv, acc[1 * 4 + j]);
      acc[2 * 4 + j] = wmma4(a2, bv, acc[2 * 4 + j]);
      acc[3 * 4 + j] = wmma4(a3, bv, acc[3 * 4 + j]);
    }
  }

#pragma unroll
  for (int ii = 0; ii < 4; ++ii) {
#pragma unroll
    for (int j = 0; j < 4; ++j) {
      int n = nbk * 64 + j * 16 + lm;
      float bb = bias[n];
#pragma unroll
      for (int v = 0; v < 8; ++v) {
        int m = mb * 64 + ii * 16 + v + 8 * kh;
        out[(size_t)m * VV + n] = acc[ii * 4 + j][v] + bb;
      }
    }
  }
}

// ---------------------------------------------------------------------------
extern "C" void kernel_launch(void* const* d_in, const int* in_sizes, int n_in,
                              void* d_out, int out_size, void* d_ws, size_t ws_size,
                              hipStream_t stream) {
  (void)in_sizes; (void)n_in; (void)out_size; (void)ws_size;
  const int* enc_in = (const int*)d_in[0];
  const int* dec_in = (const int*)d_in[1];
  const float* enc_emb = (const float*)d_in[2];
  const float* enc_Wih = (const float*)d_in[3];
  const float* enc_Whh = (const float*)d_in[4];
  const float* enc_bih = (const float*)d_in[5];
  const float* enc_bhh = (const float*)d_in[6];
  const float* dec_emb = (const float*)d_in[7];
  const float* dec_Wih = (const float*)d_in[8];
  const float* dec_Whh = (const float*)d_in[9];
  const float* dec_bih = (const float*)d_in[10];
  const float* dec_bhh = (const float*)d_in[11];
  const float* attn_scale = (const float*)d_in[12];
  const float* out_W = (const float*)d_in[13];
  const float* out_b = (const float*)d_in[14];
  float* logits = (float*)d_out;
  float* ws = (float*)d_ws;

  // workspace layout (float offsets)
  const size_t o_encx = 0;                                  // S*B*E
  const size_t o_decx = o_encx + (size_t)SS * BB * EE;      // T*B*E
  const size_t o_xge  = o_decx + (size_t)TT * BB * EE;      // S*B*4H
  const size_t o_xgd  = o_xge + (size_t)SS * BB * GG;       // T*B*4H
  const size_t o_encout = o_xgd + (size_t)TT * BB * GG;     // B*S*H
  const size_t o_decout = o_encout + (size_t)BB * SS * HH;  // B*T*H
  const size_t o_concat = o_decout + (size_t)BB * TT * HH;  // B*T*2H
  const size_t o_h = o_concat + (size_t)BB * TT * 2 * HH;   // 2*16*H
  const size_t o_c = o_h + (size_t)2 * 16 * HH;             // 16*H
  const size_t o_cnt = o_c + (size_t)16 * HH;               // 2 uints
  unsigned* cnt = (unsigned*)(ws + o_cnt);

  const int hc_n = 2 * 16 * HH + 16 * HH;  // 24576 floats
  k_init<<<(hc_n + 255) / 256, 256, 0, stream>>>(ws + o_h, hc_n, cnt);

  const int gather_blocks = (SS * BB * EE) / 256;  // 2048
  k_gather<<<gather_blocks, 256, 0, stream>>>(enc_in, enc_emb, ws + o_encx, SS);
  k_gather<<<gather_blocks, 256, 0, stream>>>(dec_in, dec_emb, ws + o_decx, TT);

  // Xg GEMMs: 2048 waves each -> 512 blocks of 128 threads
  k_xg_gemm<<<512, 128, 0, stream>>>(ws + o_encx, enc_Wih, enc_bih, enc_bhh, ws + o_xge);
  k_xg_gemm<<<512, 128, 0, stream>>>(ws + o_decx, dec_Wih, dec_bih, dec_bhh, ws + o_xgd);

  // Recurrences: persistent 8 blocks x 128 threads (32 waves, grid-synced)
  k_lstm<<<8, 128, 0, stream>>>(ws + o_xge, enc_Whh, ws + o_h, ws + o_c,
                                ws + o_encout, cnt + 0, SS);
  k_lstm<<<8, 128, 0, stream>>>(ws + o_xgd, dec_Whh, ws + o_h, ws + o_c,
                                ws + o_decout, cnt + 1, TT);

  // Attention + concat
  k_attn<<<BB * TT, 128, 0, stream>>>(ws + o_encout, ws + o_decout, attn_scale,
                                      ws + o_concat);

  // Output projection: 8000 waves -> 2000 blocks of 128 threads
  k_out_gemm<<<2000, 128, 0, stream>>>(ws + o_concat, out_W, out_b, logits);
}